// GNN_Layer_Init_57217554317353
// MI455X (gfx1250) — compile-verified
//
#include <hip/hip_runtime.h>
#include <stdint.h>

typedef __attribute__((ext_vector_type(2))) float v2f;
typedef __attribute__((ext_vector_type(8))) float v8f;

#define FEAT 128            // OUT_FEATURES (compile-time: 8 waves * 16 feats)
#define ROWS_PER_BIN 16
#define EDGE_CHUNK 1024     // edges staged in LDS per chunk (8 KB)

// ---------------------------------------------------------------------------
// Pass 0: zero bin counters and scatter cursors
// ---------------------------------------------------------------------------
__global__ void gnn_zero_kernel(int* __restrict__ counts, int* __restrict__ cursor, int nb) {
    int i = blockIdx.x * blockDim.x + threadIdx.x;
    if (i < nb) { counts[i] = 0; cursor[i] = 0; }
}

// ---------------------------------------------------------------------------
// Pass 1: histogram edges by destination row block (row >> 4)
// ---------------------------------------------------------------------------
__global__ void gnn_count_kernel(const int* __restrict__ edge_row,
                                 int* __restrict__ counts, int E) {
    int e = blockIdx.x * blockDim.x + threadIdx.x;
    if (e < E) atomicAdd(&counts[edge_row[e] >> 4], 1);
}

// ---------------------------------------------------------------------------
// Pass 2: exclusive scan of bin counts (single workgroup; nb ~ 6250)
// ---------------------------------------------------------------------------
__global__ void gnn_scan_kernel(const int* __restrict__ counts,
                                int* __restrict__ offsets, int nb) {
    __shared__ int partial[256];
    __shared__ int partial_pre[256];
    const int t = threadIdx.x;
    const int chunk = (nb + 255) / 256;
    const int beg = t * chunk;
    const int end = min(beg + chunk, nb);
    int s = 0;
    for (int i = beg; i < end; ++i) s += counts[i];
    partial[t] = s;
    __syncthreads();
    if (t == 0) {
        int run = 0;
        for (int i = 0; i < 256; ++i) { partial_pre[i] = run; run += partial[i]; }
        offsets[nb] = run;
    }
    __syncthreads();
    int run = partial_pre[t];
    for (int i = beg; i < end; ++i) { offsets[i] = run; run += counts[i]; }
}

// ---------------------------------------------------------------------------
// Pass 3: scatter edges into bin-contiguous storage.
// Packed edge: .x = (row & 15) << 28 | col (col < 2^28), .y = bits(val)
// ---------------------------------------------------------------------------
__global__ void gnn_scatter_kernel(const int* __restrict__ edge_row,
                                   const int* __restrict__ edge_col,
                                   const float* __restrict__ edge_val,
                                   const int* __restrict__ offsets,
                                   int* __restrict__ cursor,
                                   uint2* __restrict__ binned, int E) {
    int e = blockIdx.x * blockDim.x + threadIdx.x;
    if (e < E) {
        const int r = edge_row[e];
        const int bin = r >> 4;
        const int pos = offsets[bin] + atomicAdd(&cursor[bin], 1);
        uint2 pk;
        pk.x = ((unsigned)edge_col[e] & 0x0FFFFFFFu) | (((unsigned)r & 15u) << 28);
        pk.y = __float_as_uint(edge_val[e]);
        binned[pos] = pk;
    }
}

// ---------------------------------------------------------------------------
// Pass 4: per-row-block SpMM via V_WMMA_F32_16X16X4_F32.
// One workgroup (8 wave32) per block of 16 destination rows; wave w owns
// features [16w, 16w+16). Edges are cooperatively staged into LDS (zero-
// padded to a multiple of 4) so the compute loop is branch-free:
//   A[16x4]: A[m,k] = (row_k == m) ? val_k : 0   (selector matrix)
//   B[4x16]: weight[col_k, featBase + n]          (gathered slice)
// f32 operand layout (ISA 7.12.2): lanes 0-15 hold K=0 (V0) / K=1 (V1),
// lanes 16-31 hold K=2 (V0) / K=3 (V1). The two edges a half-wave needs per
// group are 16 contiguous LDS bytes. The loop is software-pipelined one
// group deep: LDS edge read + L2 weight gathers for group g+1 issue before
// the WMMA of group g, so the WMMA never drains loadcnt to zero.
// C/D VGPR r: lanes 0-15 -> M=r, lanes 16-31 -> M=r+8. Output written once,
// fused with the bias add.
// ---------------------------------------------------------------------------
__global__ void __launch_bounds__(256)
gnn_spmm_wmma_kernel(const uint2* __restrict__ binned,
                     const int* __restrict__ offsets,
                     const int* __restrict__ counts,
                     const float* __restrict__ weight,
                     const float* __restrict__ bias,
                     float* __restrict__ out,
                     int n_nodes) {
    __shared__ uint2 s_edges[EDGE_CHUNK];

    const int bin      = blockIdx.x;
    const int lane     = threadIdx.x & 31;
    const int wave     = threadIdx.x >> 5;
    const int featBase = wave << 4;          // 8 waves x 16 features = 128
    const int laneN    = lane & 15;          // N (and M) index within half-wave
    const int half     = lane >> 4;          // 0: K=0,1   1: K=2,3

    const int beg    = offsets[bin];
    const int nEdges = counts[bin];
    const int colF   = featBase + laneN;
    const float bv   = bias[colF];           // early, overlaps with staging

    v8f acc = {};

    for (int chunk = 0; chunk < nEdges; chunk += EDGE_CHUNK) {
        const int cnt    = min(EDGE_CHUNK, nEdges - chunk);
        const int padded = (cnt + 3) & ~3;   // zero-pad to whole 4-edge groups

        __syncthreads();                     // previous chunk fully consumed
        for (int i = threadIdx.x; i < padded; i += 256)
            s_edges[i] = (i < cnt) ? binned[beg + chunk + i] : make_uint2(0u, 0u);
        __syncthreads();

        const int rounds = padded >> 2;      // >= 1 whenever we get here

        // ---- pipeline prologue: fetch group 0 ----
        uint2 ea = s_edges[(half << 1) + 0];
        uint2 eb = s_edges[(half << 1) + 1];
        float w0 = weight[(ea.x & 0x0FFFFFFFu) * FEAT + colF];
        float w1 = weight[(eb.x & 0x0FFFFFFFu) * FEAT + colF];

        // ---- steady state: prefetch g+1, WMMA g (branch-free body) ----
#pragma unroll 2
        for (int g = 0; g < rounds - 1; ++g) {
            const unsigned rowA = ea.x >> 28, rowB = eb.x >> 28;
            const float    va   = __uint_as_float(ea.y);
            const float    vb   = __uint_as_float(eb.y);
            v2f a, b;
            a[0] = (rowA == (unsigned)laneN) ? va : 0.0f;
            a[1] = (rowB == (unsigned)laneN) ? vb : 0.0f;
            b[0] = w0;
            b[1] = w1;

            // prefetch group g+1 (independent of this group's WMMA)
            ea = s_edges[((g + 1) << 2) + (half << 1) + 0];
            eb = s_edges[((g + 1) << 2) + (half << 1) + 1];
            w0 = weight[(ea.x & 0x0FFFFFFFu) * FEAT + colF];
            w1 = weight[(eb.x & 0x0FFFFFFFu) * FEAT + colF];

            acc = __builtin_amdgcn_wmma_f32_16x16x4_f32(
                false, a, false, b, (short)0, acc, false, false);
        }

        // ---- pipeline epilogue: last group ----
        {
            const unsigned rowA = ea.x >> 28, rowB = eb.x >> 28;
            v2f a, b;
            a[0] = (rowA == (unsigned)laneN) ? __uint_as_float(ea.y) : 0.0f;
            a[1] = (rowB == (unsigned)laneN) ? __uint_as_float(eb.y) : 0.0f;
            b[0] = w0;
            b[1] = w1;
            acc = __builtin_amdgcn_wmma_f32_16x16x4_f32(
                false, a, false, b, (short)0, acc, false, false);
        }
    }

    // Epilogue: single store per output element, fused bias add; bounds
    // check hoisted out of the store loop (only the last bin can be partial).
    const int baseRow = bin * ROWS_PER_BIN + half * 8;
    float* outp = out + (size_t)baseRow * FEAT + colF;
    if (baseRow + 8 <= n_nodes) {
#pragma unroll
        for (int r = 0; r < 8; ++r)
            outp[(size_t)r * FEAT] = acc[r] + bv;
    } else {
#pragma unroll
        for (int r = 0; r < 8; ++r)
            if (baseRow + r < n_nodes)
                outp[(size_t)r * FEAT] = acc[r] + bv;
    }
}

// ---------------------------------------------------------------------------
// Host-side orchestration (all on `stream`, graph-capture safe)
// ---------------------------------------------------------------------------
extern "C" void kernel_launch(void* const* d_in, const int* in_sizes, int n_in,
                              void* d_out, int out_size, void* d_ws, size_t ws_size,
                              hipStream_t stream) {
    const int*   edge_row = (const int*)d_in[0];
    const int*   edge_col = (const int*)d_in[1];
    const float* edge_val = (const float*)d_in[2];
    const float* weight   = (const float*)d_in[3];
    const float* bias     = (const float*)d_in[4];
    float*       out      = (float*)d_out;

    const int E       = in_sizes[0];
    const int F       = in_sizes[4];                 // 128
    const int n_nodes = in_sizes[3] / F;             // 100000
    const int nb      = (n_nodes + ROWS_PER_BIN - 1) / ROWS_PER_BIN;  // 6250

    // Workspace carve-up: counts[nb] | cursor[nb] | offsets[nb+1] | binned[E]
    char*  ws  = (char*)d_ws;
    size_t off = 0;
    int* counts  = (int*)(ws + off); off += (size_t)nb * sizeof(int);
    int* cursor  = (int*)(ws + off); off += (size_t)nb * sizeof(int);
    int* offsets = (int*)(ws + off); off += (size_t)(nb + 1) * sizeof(int);
    off = (off + 15) & ~(size_t)15;
    uint2* binned = (uint2*)(ws + off);              // E * 8 bytes (~25.6 MB)

    const int T = 256;
    gnn_zero_kernel<<<(nb + T - 1) / T, T, 0, stream>>>(counts, cursor, nb);
    gnn_count_kernel<<<(E + T - 1) / T, T, 0, stream>>>(edge_row, counts, E);
    gnn_scan_kernel<<<1, T, 0, stream>>>(counts, offsets, nb);
    gnn_scatter_kernel<<<(E + T - 1) / T, T, 0, stream>>>(
        edge_row, edge_col, edge_val, offsets, cursor, binned, E);
    gnn_spmm_wmma_kernel<<<nb, T, 0, stream>>>(
        binned, offsets, counts, weight, bias, out, n_nodes);
}